// HungarianMatcher_16990890623452
// MI455X (gfx1250) — compile-verified
//
#include <hip/hip_runtime.h>
#include <hip/hip_bf16.h>
#include <stdint.h>

// Batched Hungarian (Jonker-Volgenant shortest augmenting path), one wave32
// per batch element. bs=128, n=256. Cost tile (256KB) staged into the WGP's
// 320KB LDS via the CDNA5 Tensor Data Mover; all solver state at LDS/register
// latency; cross-lane argmin via wave32 shuffles (no barriers anywhere).

#define HM_N    256
#define HM_PER  8            // columns per lane = N / 32
#define HM_INF  1.0e9f

typedef __attribute__((ext_vector_type(4))) unsigned int hm_u32x4;
typedef __attribute__((ext_vector_type(8))) int          hm_i32x8;
typedef __attribute__((ext_vector_type(4))) int          hm_i32x4;

__global__ __launch_bounds__(32)
void hungarian_jv_kernel(const float* __restrict__ costs,
                         long long* __restrict__ out, int bs)
{
    const int b    = blockIdx.x;          // batch element
    const int lane = threadIdx.x;         // 0..31 (wave32)

    extern __shared__ char smem_raw[];
    float* costL   = (float*)smem_raw;                          // 256*256 f32 = 262144 B
    float* u_lds   = (float*)(smem_raw + 262144);               // 256 f32  (row duals)
    int*   p_lds   = (int*)  (smem_raw + 262144 + 1024);        // 256 i32  (p[j] = row matched to col j, -1 free)
    int*   way_lds = (int*)  (smem_raw + 262144 + 2048);        // 256 i32  (predecessor column on path)

    // ---------------- Stage full 256x256 cost tile into LDS (TDM) ----------------
    const unsigned long long gaddr =
        (unsigned long long)(uintptr_t)(const void*)(costs + (size_t)b * (HM_N * HM_N));
    const unsigned lds_off = (unsigned)(uintptr_t)(void*)costL; // low 32b of flat addr = LDS byte offset

#if __has_builtin(__builtin_amdgcn_tensor_load_to_lds)
    // D# group 0 (128b): [1:0] count=1 valid | [63:32] lds_addr |
    //                    [120:64] global_addr | [127:126] type=2
    hm_u32x4 g0 = {
        1u,
        lds_off,
        (unsigned)(gaddr & 0xFFFFFFFFu),
        ((unsigned)((gaddr >> 32) & 0x01FFFFFFu)) | (2u << 30)
    };
    // D# group 1 (256b): data_size=2(4B) | tensor_dim0=256 | tensor_dim1=256 |
    //                    tile_dim0=256 | tile_dim1=256 | tensor_dim0_stride=256
    hm_i32x8 g1 = {
        (int)(2u << 16),          // [17:16] data_size = 4 bytes; workgroup_mask=0
        (int)(256u << 16),        // [63:48]  tensor_dim0[15:0] = 256
        (int)(256u << 16),        // [79:64]  tensor_dim0 hi = 0; tensor_dim1[15:0] = 256
        (int)(256u << 16),        // [127:112] tile_dim0 = 256 (dim1 hi = 0)
        256,                      // [143:128] tile_dim1 = 256; tile_dim2 = 0
        256,                      // [191:160] tensor_dim0_stride[31:0] = 256 elements
        0, 0
    };
    hm_i32x4 gz4 = {0, 0, 0, 0};                 // D# groups 2/3: unused (2D tensor)
    hm_i32x8 gz8 = {0, 0, 0, 0, 0, 0, 0, 0};     // trailing group (6-arg builtin form)
    __builtin_amdgcn_tensor_load_to_lds(g0, g1, gz4, gz4, gz8, 0);
    __builtin_amdgcn_s_wait_tensorcnt(0);
#else
    // Fallback: plain cooperative copy (single wave)
    for (int t = lane; t < HM_N * HM_N; t += 32)
        costL[t] = costs[(size_t)b * (HM_N * HM_N) + t];
#endif

    // ---------------- Init duals / matching ----------------
    float vreg[HM_PER];                   // column duals, lane-owned (j = k*32+lane)
#pragma unroll
    for (int k = 0; k < HM_PER; ++k) {
        const int j = k * 32 + lane;
        u_lds[j] = 0.0f;
        p_lds[j] = -1;
        vreg[k]  = 0.0f;
    }

    // ---------------- Solve rows sequentially ----------------
    for (int i = 0; i < HM_N; ++i) {
        float minv[HM_PER];
#pragma unroll
        for (int k = 0; k < HM_PER; ++k) minv[k] = HM_INF;
        unsigned usedm = 0;               // bit k: column k*32+lane is on the tree
        int i0 = i, j0 = -1, j1 = -1;

        for (;;) {
            const float ui0 = u_lds[i0];  // wave-uniform LDS read

            // reduced costs of row i0; improve minv / record predecessor
#pragma unroll
            for (int k = 0; k < HM_PER; ++k) {
                const int j = k * 32 + lane;
                if (!((usedm >> k) & 1u)) {
                    const float cur = costL[i0 * HM_N + j] - ui0 - vreg[k];
                    if (cur < minv[k]) { minv[k] = cur; way_lds[j] = j0; }
                }
            }

            // masked argmin over 256 columns (first-index tie-break = jnp.argmin)
            float mloc = HM_INF; int jloc = 0x7fffffff;
#pragma unroll
            for (int k = 0; k < HM_PER; ++k) {
                const int j = k * 32 + lane;
                const float m = ((usedm >> k) & 1u) ? HM_INF : minv[k];
                if (m < mloc) { mloc = m; jloc = j; }
            }
#pragma unroll
            for (int off = 16; off >= 1; off >>= 1) {
                const float om = __shfl_xor(mloc, off, 32);
                const int   oj = __shfl_xor(jloc, off, 32);
                if (om < mloc || (om == mloc && oj < jloc)) { mloc = om; jloc = oj; }
            }
            const float delta = mloc;
            j1 = jloc;

            // dual updates (i unmatched; matched rows of used cols are distinct)
            if (lane == 0) u_lds[i] = u_lds[i] + delta;
#pragma unroll
            for (int k = 0; k < HM_PER; ++k) {
                const int j = k * 32 + lane;
                if ((usedm >> k) & 1u) {
                    const int pj = p_lds[j];
                    u_lds[pj] = u_lds[pj] + delta;
                    vreg[k] -= delta;
                } else {
                    minv[k] -= delta;
                }
            }
            if (lane == (j1 & 31)) usedm |= 1u << (j1 >> 5);

            const int nxt = p_lds[j1];    // wave-uniform
            if (nxt < 0) break;           // reached a free column
            i0 = nxt;
            j0 = j1;
        }

        // augment backwards along way[] from the free column j1
        if (lane == 0) {
            int jc = j1;
            while (jc >= 0) {
                const int jprev = way_lds[jc];
                p_lds[jc] = (jprev < 0) ? i : p_lds[jprev];
                jc = jprev;
            }
        }
        // per-wave LDS ordering: lane0's stores visible to next row-solve
    }

    // ---------------- Emit (rows, cols) as int64, concatenated ----------------
    long long* rows = out;                          // [bs, 256]
    long long* cols = out + (size_t)bs * HM_N;      // [bs, 256]
#pragma unroll
    for (int k = 0; k < HM_PER; ++k) {
        const int j = k * 32 + lane;
        rows[(size_t)b * HM_N + j] = (long long)j;  // arange per batch
        const int r = p_lds[j];                     // row matched to column j
        cols[(size_t)b * HM_N + r] = (long long)j;  // col_of_row[r] = j
    }
}

extern "C" void kernel_launch(void* const* d_in, const int* in_sizes, int n_in,
                              void* d_out, int out_size, void* d_ws, size_t ws_size,
                              hipStream_t stream) {
    const float* costs = (const float*)d_in[0];
    long long*   out   = (long long*)d_out;
    const int bs = in_sizes[0] / (HM_N * HM_N);     // 128

    const size_t lds_bytes = 262144 + 3 * 1024;     // cost tile + u/p/way
    (void)hipFuncSetAttribute((const void*)hungarian_jv_kernel,
                              hipFuncAttributeMaxDynamicSharedMemorySize,
                              (int)lds_bytes);

    hungarian_jv_kernel<<<bs, 32, lds_bytes, stream>>>(costs, out, bs);
}